// AngularBasis_51127290691591
// MI455X (gfx1250) — compile-verified
//
#include <hip/hip_runtime.h>
#include <hip/hip_bf16.h>
#include <stdint.h>

// ---------------------------------------------------------------------------
// Compile-time normalization constants K(l,m) = sqrt((2l+1)/(4pi) *
// (l-m)!/(l+m)!) * (m ? sqrt(2) : 1).  All folded to f32 immediates.
// ---------------------------------------------------------------------------
constexpr double k_pi = 3.14159265358979323846;

constexpr double csqrt(double x) {
    double g = (x > 1.0) ? x : 1.0;
    for (int i = 0; i < 120; ++i) g = 0.5 * (g + x / g);
    return g;
}
constexpr double cfact(int n) {
    double r = 1.0;
    for (int i = 2; i <= n; ++i) r *= (double)i;
    return r;
}
constexpr float KC(int l, int m) {
    double k = csqrt((2.0 * l + 1.0) / (4.0 * k_pi) * cfact(l - m) / cfact(l + m));
    if (m != 0) k *= csqrt(2.0);
    return (float)k;
}
struct KTable { float v[16][16]; };
constexpr KTable makeK() {
    KTable t{};
    for (int l = 0; l < 16; ++l)
        for (int m = 0; m < 16; ++m)
            t.v[l][m] = (m <= l) ? KC(l, m) : 0.0f;
    return t;
}
constexpr KTable KT = makeK();

// ---------------------------------------------------------------------------
// Fast templated kernel: 1 thread = 1 point. Wave of 32 points stages its
// 32 x (LMAX+1)^2 output tile in LDS in global output order, then streams it
// with gfx1250 async LDS->global b128 stores (fully coalesced 512B/instr,
// ASYNCcnt-tracked).  This collapses the ~Sum (2l+1)^2 = 680 partial-line
// store transactions/wave of the naive strided layout to the minimal 64.
// ---------------------------------------------------------------------------
template <int LMAX>
__global__ __launch_bounds__(128) void sph_kernel(const float* __restrict__ ct,
                                                  const float* __restrict__ ph,
                                                  float* __restrict__ out,
                                                  int N) {
    constexpr int COLS  = (LMAX + 1) * (LMAX + 1);
    constexpr int WAVES = 4;
    __shared__ __align__(16) float lds[WAVES * 32 * COLS];

    const int  tid   = threadIdx.x;
    const int  lane  = tid & 31;
    const int  wave  = tid >> 5;
    const long long i     = (long long)blockIdx.x * (WAVES * 32) + tid;
    const long long tile0 = (long long)blockIdx.x * (WAVES * 32) + wave * 32;
    const bool act  = (i < N);
    // fast path needs full 32-point tile and 16B-aligned section bases
    const bool full = ((tile0 + 32) <= (long long)N) && ((N & 3) == 0);

    float x = 0.0f, p = 0.0f;
    if (act) { x = ct[i]; p = ph[i]; }

    const float s = sqrtf(fmaxf(1.0f - x * x, 0.0f));
    float sp, cp;
    sincosf(p, &sp, &cp);

    // Associated Legendre P[l][m] (Condon-Shortley), fully unrolled -> registers
    float P[LMAX + 1][LMAX + 1];
    P[0][0] = 1.0f;
#pragma unroll
    for (int m = 1; m <= LMAX; ++m)
        P[m][m] = (-(2.0f * m - 1.0f)) * s * P[m - 1][m - 1];
#pragma unroll
    for (int m = 0; m < LMAX; ++m)
        P[m + 1][m] = (2.0f * m + 1.0f) * x * P[m][m];
#pragma unroll
    for (int m = 0; m <= LMAX; ++m) {
#pragma unroll
        for (int l = m + 2; l <= LMAX; ++l)
            P[l][m] = ((2.0f * l - 1.0f) * x * P[l - 1][m]
                       - (float)(l + m - 1) * P[l - 2][m]) / (float)(l - m);
    }

    // cos(m phi), sin(m phi) via angle-addition recurrence
    float cm[LMAX + 2], sm[LMAX + 2];
    cm[0] = 1.0f; sm[0] = 0.0f;
#pragma unroll
    for (int m = 1; m <= LMAX; ++m) {
        cm[m] = cm[m - 1] * cp - sm[m - 1] * sp;
        sm[m] = sm[m - 1] * cp + cm[m - 1] * sp;
    }

    float* wt = &lds[wave * 32 * COLS];

    if (full) {
        // Stage outputs in LDS in exact global order:
        // section l at float offset 32*l^2, point 'lane' row at lane*(2l+1).
        // Lane write stride (2l+1) is odd -> conflict-free across 64 banks.
#pragma unroll
        for (int l = 0; l <= LMAX; ++l) {
            const int W   = 2 * l + 1;
            float*    sec = wt + 32 * l * l + lane * W;
#pragma unroll
            for (int mm = 0; mm < W; ++mm) {
                const int   m  = mm - l;
                const int   am = (m < 0) ? -m : m;
                float       v  = KT.v[l][am] * P[l][am];
                if (m < 0)      v *= sm[am];
                else if (m > 0) v *= cm[am];
                sec[mm] = v;
            }
        }
        // Ensure all DS writes of this wave landed before async engine reads LDS
        asm volatile("s_wait_dscnt 0x0" ::: "memory");

        // Stream the tile: per section l, 8*(2l+1) float4s, 16B-aligned,
        // copied LDS -> global by the async data path (ASYNCcnt-tracked).
#pragma unroll
        for (int l = 0; l <= LMAX; ++l) {
            const int       NF4 = 8 * (2 * l + 1);                    // float4 count
            const long long gfl = (long long)N * (l * l) + tile0 * (2 * l + 1);
            float*          lsec = wt + 32 * l * l;
#pragma unroll
            for (int j0 = 0; j0 < (NF4 + 31) / 32; ++j0) {
                const int j = lane + 32 * j0;
                if (j < NF4) {
                    const unsigned long long ga =
                        (unsigned long long)(uintptr_t)(out + gfl + 4 * (long long)j);
                    const unsigned la =
                        (unsigned)(uintptr_t)(lsec + 4 * j);          // LDS byte addr
                    asm volatile("global_store_async_from_lds_b128 %0, %1, off"
                                 :: "v"(ga), "v"(la)
                                 : "memory");
                }
            }
        }
        // Drain async stores before the wave releases its LDS
        asm volatile("s_wait_asynccnt 0x0" ::: "memory");
    } else if (act) {
        // Tail / unaligned fallback: direct scalar stores
#pragma unroll
        for (int l = 0; l <= LMAX; ++l) {
            float* dst = out + (long long)N * (l * l) + i * (2 * l + 1);
#pragma unroll
            for (int mm = 0; mm < 2 * l + 1; ++mm) {
                const int m  = mm - l;
                const int am = (m < 0) ? -m : m;
                float     v  = KT.v[l][am] * P[l][am];
                if (m < 0)      v *= sm[am];
                else if (m > 0) v *= cm[am];
                dst[mm] = v;
            }
        }
    }
}

// ---------------------------------------------------------------------------
// Generic fallback for l_max outside the templated range (runtime loops).
// ---------------------------------------------------------------------------
__global__ void sph_dyn_kernel(const float* __restrict__ ct,
                               const float* __restrict__ ph,
                               float* __restrict__ out, int N, int lmax) {
    const long long i = (long long)blockIdx.x * blockDim.x + threadIdx.x;
    if (i >= N) return;

    const float x = ct[i], p = ph[i];
    const float s = sqrtf(fmaxf(1.0f - x * x, 0.0f));
    float sp, cp;
    sincosf(p, &sp, &cp);

    float P[16][16];
    P[0][0] = 1.0f;
    for (int m = 1; m <= lmax; ++m)
        P[m][m] = (-(2.0f * m - 1.0f)) * s * P[m - 1][m - 1];
    for (int m = 0; m < lmax; ++m)
        P[m + 1][m] = (2.0f * m + 1.0f) * x * P[m][m];
    for (int m = 0; m <= lmax; ++m)
        for (int l = m + 2; l <= lmax; ++l)
            P[l][m] = ((2.0f * l - 1.0f) * x * P[l - 1][m]
                       - (float)(l + m - 1) * P[l - 2][m]) / (float)(l - m);

    float cm[16], sm[16];
    cm[0] = 1.0f; sm[0] = 0.0f;
    for (int m = 1; m <= lmax; ++m) {
        cm[m] = cm[m - 1] * cp - sm[m - 1] * sp;
        sm[m] = sm[m - 1] * cp + cm[m - 1] * sp;
    }

    for (int l = 0; l <= lmax; ++l) {
        float* dst = out + (long long)N * (l * l) + i * (2 * l + 1);
        for (int mm = 0; mm < 2 * l + 1; ++mm) {
            const int m  = mm - l;
            const int am = (m < 0) ? -m : m;
            float     v  = KT.v[l][am] * P[l][am];
            if (m < 0)      v *= sm[am];
            else if (m > 0) v *= cm[am];
            dst[mm] = v;
        }
    }
}

// ---------------------------------------------------------------------------
extern "C" void kernel_launch(void* const* d_in, const int* in_sizes, int n_in,
                              void* d_out, int out_size, void* d_ws, size_t ws_size,
                              hipStream_t stream) {
    (void)d_ws; (void)ws_size; (void)n_in;
    const float* ct  = (const float*)d_in[0];
    const float* ph  = (const float*)d_in[1];
    float*       out = (float*)d_out;
    const int    N   = in_sizes[0];
    if (N <= 0) return;

    // Recover l_max on host: out_size = N * (l_max+1)^2
    const int cols = out_size / N;
    int r = 1;
    while (r * r < cols) ++r;
    const int lmax = r - 1;

    const int BT     = 128;                 // 4 waves of 32
    const int blocks = (N + BT - 1) / BT;

    switch (lmax) {
        case 0: sph_kernel<0><<<blocks, BT, 0, stream>>>(ct, ph, out, N); break;
        case 1: sph_kernel<1><<<blocks, BT, 0, stream>>>(ct, ph, out, N); break;
        case 2: sph_kernel<2><<<blocks, BT, 0, stream>>>(ct, ph, out, N); break;
        case 3: sph_kernel<3><<<blocks, BT, 0, stream>>>(ct, ph, out, N); break;
        case 4: sph_kernel<4><<<blocks, BT, 0, stream>>>(ct, ph, out, N); break;
        case 5: sph_kernel<5><<<blocks, BT, 0, stream>>>(ct, ph, out, N); break;
        case 6: sph_kernel<6><<<blocks, BT, 0, stream>>>(ct, ph, out, N); break;
        case 7: sph_kernel<7><<<blocks, BT, 0, stream>>>(ct, ph, out, N); break;
        case 8: sph_kernel<8><<<blocks, BT, 0, stream>>>(ct, ph, out, N); break;
        default: {
            int lm = lmax > 15 ? 15 : lmax;
            sph_dyn_kernel<<<(N + 255) / 256, 256, 0, stream>>>(ct, ph, out, N, lm);
            break;
        }
    }
}